// GNN_16690242913041
// MI455X (gfx1250) — compile-verified
//
#include <hip/hip_runtime.h>

// ---------------------------------------------------------------------------
// GraphSAGE 2-layer inference for MI455X (gfx1250, wave32, WMMA).
//   agg = segment_mean(x[src] -> dst);  h = relu(agg@W1_l + b1 + x@W1_r)
//   agg2= segment_mean(h[src] -> dst);  out = agg2@W2_l + b2 + h@W2_r
// Aggregation: fp32 hardware atomics into an L2-resident buffer.
// GEMMs: V_WMMA_F32_16X16X4_F32 (full fp32 precision on the matrix pipe),
// mean-scale folded into the A-fragment load, bias+ReLU fused in epilogue.
// ---------------------------------------------------------------------------

#define N_NODES 100000
#define N_EDGES 600000
#define DIM 128

typedef __attribute__((ext_vector_type(2))) float v2f;
typedef __attribute__((ext_vector_type(8))) float v8f;

// ----------------------------- utility kernels -----------------------------

__global__ void GNN_zero_kernel(float4* __restrict__ p, int n4) {
    int i = blockIdx.x * blockDim.x + threadIdx.x;
    if (i < n4) p[i] = make_float4(0.f, 0.f, 0.f, 0.f);
}

__global__ void GNN_count_kernel(const int* __restrict__ dst,
                                 int* __restrict__ cnt) {
    int e = blockIdx.x * blockDim.x + threadIdx.x;
    if (e < N_EDGES) atomicAdd(&cnt[dst[e]], 1);
}

__global__ void GNN_invcnt_kernel(const int* __restrict__ cnt,
                                  float* __restrict__ inv) {
    int i = blockIdx.x * blockDim.x + threadIdx.x;
    if (i < N_NODES) {
        int c = cnt[i];
        inv[i] = 1.0f / (float)(c > 0 ? c : 1);
    }
}

// One edge per 32 threads: coalesced float4 gather of the 128-float source
// row, 4 hardware fp32 atomics per lane into agg[dst]. agg (51.2MB) is
// L2-resident on MI455X (192MB L2), so these atomics resolve in L2.
__global__ void GNN_scatter_kernel(const float4* __restrict__ feat4,
                                   const int* __restrict__ src,
                                   const int* __restrict__ dst,
                                   float* __restrict__ agg) {
    int t = blockIdx.x * blockDim.x + threadIdx.x;
    if (t >= N_EDGES * 32) return;
    int e = t >> 5;
    int q = t & 31;           // which float4 of the 128-float row
    int s = src[e];
    int d = dst[e];
    float4 v = feat4[(size_t)s * 32 + q];
    float* a = agg + (size_t)d * DIM + q * 4;
    unsafeAtomicAdd(a + 0, v.x);   // global_atomic_add_f32
    unsafeAtomicAdd(a + 1, v.y);
    unsafeAtomicAdd(a + 2, v.z);
    unsafeAtomicAdd(a + 3, v.w);
}

// ------------------------------- WMMA GEMM ---------------------------------
// out[16 x 128] per block: 8 waves, each wave owns one 16x16 tile.
//   D = (agg * invcnt) @ Wl + bias + xin @ Wr   (+ ReLU if RELU)
// K-loop: 32 steps of V_WMMA_F32_16X16X4_F32, two WMMAs per step.
//
// Fragment layouts (ISA 7.12.2, wave32):
//   A 16x4 f32:  lane m=lane&15 -> row M=m; half=lane>>4 -> K = k+2*half (v0), +1 (v1)
//   B  4x16 f32: v0 = B[k+2*half][col0+m], v1 = B[k+2*half+1][col0+m]
//   C/D 16x16:   VGPR j -> M = j + 8*half, N = col0+m
template <bool RELU>
__global__ __launch_bounds__(256) void GNN_sage_gemm(
    const float* __restrict__ agg, const float* __restrict__ xin,
    const float* __restrict__ invc, const float* __restrict__ Wl,
    const float* __restrict__ bias, const float* __restrict__ Wr,
    float* __restrict__ out) {
    const int wave = threadIdx.x >> 5;   // 0..7 -> col tile
    const int lane = threadIdx.x & 31;
    const int half = lane >> 4;          // 0/1
    const int m    = lane & 15;
    const int row0 = blockIdx.x * 16;    // N_NODES = 16 * 6250 exactly
    const int col0 = wave * 16;
    const int row  = row0 + m;

    const float sA = invc[row];          // fold mean division into A load
    const float* aRow = agg + (size_t)row * DIM;
    const float* xRow = xin + (size_t)row * DIM;

    v8f acc = {0.f, 0.f, 0.f, 0.f, 0.f, 0.f, 0.f, 0.f};

#pragma unroll 4
    for (int k = 0; k < DIM; k += 4) {
        const int ka = k + 2 * half;
        float2 ta = *(const float2*)(aRow + ka);   // 8B-aligned (ka even)
        float2 tx = *(const float2*)(xRow + ka);
        v2f Aagg; Aagg[0] = ta.x * sA; Aagg[1] = ta.y * sA;
        v2f Ax;   Ax[0]   = tx.x;      Ax[1]   = tx.y;
        v2f Bl;   Bl[0] = Wl[(size_t)ka * DIM + col0 + m];
                  Bl[1] = Wl[(size_t)(ka + 1) * DIM + col0 + m];
        v2f Br;   Br[0] = Wr[(size_t)ka * DIM + col0 + m];
                  Br[1] = Wr[(size_t)(ka + 1) * DIM + col0 + m];
        // (neg_a, A, neg_b, B, c_mod, C, reuse_a, reuse_b)
        acc = __builtin_amdgcn_wmma_f32_16x16x4_f32(
            false, Aagg, false, Bl, (short)0, acc, false, false);
        acc = __builtin_amdgcn_wmma_f32_16x16x4_f32(
            false, Ax, false, Br, (short)0, acc, false, false);
    }

    const float bv = bias[col0 + m];
#pragma unroll
    for (int j = 0; j < 8; ++j) {
        float v = acc[j] + bv;
        if (RELU) v = fmaxf(v, 0.0f);
        out[(size_t)(row0 + j + 8 * half) * DIM + col0 + m] = v;
    }
}

// ------------------------------- launcher ----------------------------------

extern "C" void kernel_launch(void* const* d_in, const int* in_sizes, int n_in,
                              void* d_out, int out_size, void* d_ws,
                              size_t ws_size, hipStream_t stream) {
    const float* x   = (const float*)d_in[0];
    const int*   ei  = (const int*)d_in[1];   // [2, E] row-major
    const float* W1l = (const float*)d_in[2];
    const float* b1l = (const float*)d_in[3];
    const float* W1r = (const float*)d_in[4];
    const float* W2l = (const float*)d_in[5];
    const float* b2l = (const float*)d_in[6];
    const float* W2r = (const float*)d_in[7];
    float* out = (float*)d_out;

    // workspace layout: agg[N*128 f32] | h[N*128 f32] | cnt[N i32] | inv[N f32]
    float* agg = (float*)d_ws;
    float* h   = agg + (size_t)N_NODES * DIM;
    int*   cnt = (int*)(h + (size_t)N_NODES * DIM);
    float* inv = (float*)(cnt + N_NODES);

    const int* src = ei;
    const int* dst = ei + N_EDGES;

    const int NF4   = N_NODES * DIM / 4;   // float4 count of agg
    const int CNT4  = N_NODES / 4;         // float4 count of cnt
    const int SCT_T = N_EDGES * 32;        // scatter threads

    // degrees (same graph both layers)
    GNN_zero_kernel<<<(CNT4 + 255) / 256, 256, 0, stream>>>((float4*)cnt, CNT4);
    GNN_count_kernel<<<(N_EDGES + 255) / 256, 256, 0, stream>>>(dst, cnt);
    GNN_invcnt_kernel<<<(N_NODES + 255) / 256, 256, 0, stream>>>(cnt, inv);

    // layer 1: agg = sum x[src] -> dst ; h = relu(mean@W1l + b1 + x@W1r)
    GNN_zero_kernel<<<(NF4 + 255) / 256, 256, 0, stream>>>((float4*)agg, NF4);
    GNN_scatter_kernel<<<(SCT_T + 255) / 256, 256, 0, stream>>>(
        (const float4*)x, src, dst, agg);
    GNN_sage_gemm<true><<<N_NODES / 16, 256, 0, stream>>>(
        agg, x, inv, W1l, b1l, W1r, h);

    // layer 2: agg = sum h[src] -> dst ; out = mean@W2l + b2 + h@W2r
    GNN_zero_kernel<<<(NF4 + 255) / 256, 256, 0, stream>>>((float4*)agg, NF4);
    GNN_scatter_kernel<<<(SCT_T + 255) / 256, 256, 0, stream>>>(
        (const float4*)h, src, dst, agg);
    GNN_sage_gemm<false><<<N_NODES / 16, 256, 0, stream>>>(
        agg, h, inv, W2l, b2l, W2r, out);
}